// MDRNN_22892175687946
// MI455X (gfx1250) — compile-verified
//
#include <hip/hip_runtime.h>
#include <math.h>

// ---------------------------------------------------------------------------
// Vanilla RNN on MI455X (gfx1250, wave32, WMMA, async global->LDS):
//   xp = x @ wax + ba            (bandwidth-bound big GEMM -> kernel 1)
//   a_t = tanh(a_{t-1}@waa+xp_t) (sequential scan, barrier-bound -> kernel 2)
// f32 WMMA (16x16x4): the scan is latency/barrier-bound, so fp32 fidelity of
// the 1024-step recurrence is free; kernel 1 is HBM-bound (~256MB @ 23.3TB/s).
// Staging uses GLOBAL_LOAD_ASYNC_TO_LDS_B128 (ASYNCcnt), no VGPR bounce.
// ---------------------------------------------------------------------------

typedef float v2f __attribute__((ext_vector_type(2)));
typedef float v8f __attribute__((ext_vector_type(8)));
typedef int v4i __attribute__((ext_vector_type(4)));

#define B_DIM 32
#define T_DIM 1024
#define D_DIM 1024
#define U_DIM 1024
#define NBLK_SCAN 32

// ---- async global->LDS copy (16B), with synchronous fallback ---------------
#if defined(__has_builtin)
#if __has_builtin(__builtin_amdgcn_global_load_async_to_lds_b128) && \
    __has_builtin(__builtin_amdgcn_s_wait_asynccnt)
#define HAVE_ASYNC_LDS 1
#endif
#endif

#if defined(HAVE_ASYNC_LDS)
// Prototype (from clang): (int4 as1* src, int4 as3* dst, imm offset, imm cpol)
#define ASYNC_CP16(dst_lds, src_glb)                                  \
  __builtin_amdgcn_global_load_async_to_lds_b128(                     \
      (__attribute__((address_space(1))) v4i*)(void*)(src_glb),       \
      (__attribute__((address_space(3))) v4i*)(void*)(dst_lds), 0, 0)
#define ASYNC_WAIT() __builtin_amdgcn_s_wait_asynccnt(0)
#else
#define ASYNC_CP16(dst_lds, src_glb) \
  (*(float4*)(dst_lds) = *(const float4*)(src_glb))
#define ASYNC_WAIT() ((void)0)
#endif

// ---------------------------------------------------------------------------
// Kernel 1: out[b,t,u] = sum_d x[b,t,d] * wax[d,u] + ba[u]
// Block: M=32 x N=64 macro tile, 8 waves, one 16x16 WMMA tile per wave.
// K chunks of 32 streamed to LDS with async b128 copies, double buffered.
// ---------------------------------------------------------------------------
__global__ __launch_bounds__(256) void xp_gemm_kernel(
    const float* __restrict__ x, const float* __restrict__ wax,
    const float* __restrict__ ba, float* __restrict__ out) {
  constexpr int KC = 32;
  constexpr int AS_S = 36;  // 16B rows; banks 4*(9r mod 16): conflict-free b64
  constexpr int BS_S = 72;  // 16B rows; upper half-wave lands +16 banks
  constexpr int NCHUNK = D_DIM / KC;
  __shared__ __align__(16) float As[2][32 * AS_S];
  __shared__ __align__(16) float Bs[2][KC * BS_S];

  const int m0 = blockIdx.x * 32;  // 1024 M-blocks over B*T rows
  const int n0 = blockIdx.y * 64;  // 16 N-blocks over U cols
  const int tid = threadIdx.x;
  const int wave = tid >> 5, lane = tid & 31;
  const int mi = wave & 1;   // 2 M-tiles
  const int ni = wave >> 1;  // 4 N-tiles
  const int lrow = lane & 15, lhi = lane >> 4;

  const float* xg = x + (size_t)m0 * D_DIM;
  const float* wg = wax + n0;
  // per-thread staging assignment: 1 A b128 + 2 B b128
  const int ar = tid >> 3, ac = (tid & 7) * 4;     // A: 32x32 = 256 xfers
  const int br0 = tid >> 4, bc0 = (tid & 15) * 4;  // B: 32x64 = 512 xfers
  const int br1 = br0 + 16, bc1 = bc0;

  auto issue = [&](int buf, int k0) {
    ASYNC_CP16(&As[buf][ar * AS_S + ac], &xg[(size_t)ar * D_DIM + k0 + ac]);
    ASYNC_CP16(&Bs[buf][br0 * BS_S + bc0],
               &wg[(size_t)(k0 + br0) * U_DIM + bc0]);
    ASYNC_CP16(&Bs[buf][br1 * BS_S + bc1],
               &wg[(size_t)(k0 + br1) * U_DIM + bc1]);
  };

  issue(0, 0);
  v8f acc = {};
  for (int i = 0; i < NCHUNK; ++i) {
    ASYNC_WAIT();     // this thread's async copies for buf[i&1] complete
    __syncthreads();  // everyone's copies visible; prev buf free for reuse
    if (i + 1 < NCHUNK) issue((i + 1) & 1, (i + 1) * KC);
    const float* Ab = As[i & 1];
    const float* Bb = Bs[i & 1];
#pragma unroll
    for (int kk = 0; kk < KC; kk += 4) {
      // A 16x4 frag: lane = (row = lane&15, k = kk + 2*(lane>>4) + {0,1})
      v2f a = *(const v2f*)&Ab[(mi * 16 + lrow) * AS_S + kk + 2 * lhi];
      // B 4x16 frag: VGPR j holds B[kk + 2*(lane>>4) + j][lane&15]
      v2f b;
      b.x = Bb[(kk + 2 * lhi) * BS_S + ni * 16 + lrow];
      b.y = Bb[(kk + 2 * lhi + 1) * BS_S + ni * 16 + lrow];
      acc = __builtin_amdgcn_wmma_f32_16x16x4_f32(false, a, false, b,
                                                  (short)0, acc, false, false);
    }
    __syncthreads();  // done reading buf[i&1] before it is refilled at i+2
  }
  // Epilogue: C layout -> VGPR i, lanes 0-15: M=i; lanes 16-31: M=8+i
  const int col = n0 + ni * 16 + lrow;
  const float bias = ba[col];
#pragma unroll
  for (int i = 0; i < 8; ++i) {
    const int row = m0 + mi * 16 + i + 8 * lhi;
    out[(size_t)row * U_DIM + col] = acc[i] + bias;
  }
}

// ---------------------------------------------------------------------------
// Device-wide sense barrier in d_ws.
// ---------------------------------------------------------------------------
__device__ __forceinline__ void grid_barrier(unsigned* cnt, unsigned* gen,
                                             unsigned nblk) {
  __threadfence();  // release this block's global stores
  __syncthreads();
  if (threadIdx.x == 0) {
    const unsigned g =
        __hip_atomic_load(gen, __ATOMIC_RELAXED, __HIP_MEMORY_SCOPE_AGENT);
    const unsigned arrived = __hip_atomic_fetch_add(
        cnt, 1u, __ATOMIC_ACQ_REL, __HIP_MEMORY_SCOPE_AGENT);
    if (arrived == nblk - 1) {
      __hip_atomic_store(cnt, 0u, __ATOMIC_RELAXED, __HIP_MEMORY_SCOPE_AGENT);
      __hip_atomic_fetch_add(gen, 1u, __ATOMIC_RELEASE,
                             __HIP_MEMORY_SCOPE_AGENT);
    } else {
      while (__hip_atomic_load(gen, __ATOMIC_ACQUIRE,
                               __HIP_MEMORY_SCOPE_AGENT) == g) {
        __builtin_amdgcn_s_sleep(1);  // 64 clk poll: barrier is on crit. path
      }
    }
  }
  __syncthreads();
  __threadfence();  // acquire before re-reading other blocks' data
}

// ---------------------------------------------------------------------------
// Kernel 2: persistent scan. 32 blocks x 128 threads (4 waves).
// Block owns 32 N-columns; waa slice [1024x32] (160 KB padded) resident in
// LDS for the whole kernel (320 KB WGP LDS). Per step: async-stage a_{t-1}
// (from `out`, which is the natural double-buffered exchange) AND the block's
// xp_t slice, one ASYNCcnt wait, 256 f32 WMMAs, tanh epilogue, grid barrier.
// ---------------------------------------------------------------------------
__global__ __launch_bounds__(128) void rnn_scan_kernel(
    const float* __restrict__ waa, float* __restrict__ out,
    unsigned* __restrict__ bar) {
  constexpr int WS_S = 40;    // 32+8 pad: upper half-wave -> +16 banks
  constexpr int AS_S = 1028;  // 1024+4 pad: 4r mod 64 banks, 16B rows
  constexpr int XS_S = 36;    // xp tile rows, 16B aligned
  extern __shared__ __align__(16) float sm[];
  float* Ws = sm;                 // U_DIM * WS_S
  float* As = Ws + U_DIM * WS_S;  // 32 * AS_S
  float* Xs = As + 32 * AS_S;     // 32 * XS_S

  const int tid = threadIdx.x;
  const int wave = tid >> 5, lane = tid & 31;
  const int mi = wave & 1;   // batch halves 0-15 / 16-31
  const int ni = wave >> 1;  // 2 N-tiles of the block's 32 columns
  const int lrow = lane & 15, lhi = lane >> 4;
  const int n0 = blockIdx.x * 32;

  // Preload persistent waa slice: Ws[k][n] = waa[k, n0+n]
  for (int i = tid; i < U_DIM * 32; i += 128) {
    const int k = i >> 5, n = i & 31;
    Ws[k * WS_S + n] = waa[(size_t)k * U_DIM + n0 + n];
  }
  // a_{-1} = 0
  for (int i = tid; i < 32 * AS_S; i += 128) As[i] = 0.0f;
  __syncthreads();

  unsigned* cnt = bar;
  unsigned* gen = bar + 32;

  for (int t = 0; t < T_DIM; ++t) {
    if (t > 0) {
      // Async-stage a_{t-1}: 32x1024 f32 = 8192 b128, 64 per thread
      for (int i = tid; i < 32 * (U_DIM / 4); i += 128) {
        const int b = i >> 8, k4 = (i & 255) * 4;
        ASYNC_CP16(&As[b * AS_S + k4],
                   &out[((size_t)b * T_DIM + (t - 1)) * U_DIM + k4]);
      }
    }
    // Async-prefetch this block's xp_t slice: 32x32 f32 = 256 b128, 2/thread
#pragma unroll
    for (int j = 0; j < 2; ++j) {
      const int i = tid + j * 128;
      const int b = i >> 3, c4 = (i & 7) * 4;
      ASYNC_CP16(&Xs[b * XS_S + c4],
                 &out[((size_t)b * T_DIM + t) * U_DIM + n0 + c4]);
    }
    ASYNC_WAIT();
    __syncthreads();

    v8f acc = {};
#pragma unroll 8
    for (int k = 0; k < U_DIM; k += 4) {
      v2f a = *(const v2f*)&As[(mi * 16 + lrow) * AS_S + k + 2 * lhi];
      v2f b;
      b.x = Ws[(k + 2 * lhi) * WS_S + ni * 16 + lrow];
      b.y = Ws[(k + 2 * lhi + 1) * WS_S + ni * 16 + lrow];
      acc = __builtin_amdgcn_wmma_f32_16x16x4_f32(false, a, false, b,
                                                  (short)0, acc, false, false);
    }

    // a_t = tanh(acc + xp_t) written in place over xp_t
    const int col = ni * 16 + lrow;
#pragma unroll
    for (int i = 0; i < 8; ++i) {
      const int b = mi * 16 + i + 8 * lhi;
      const size_t idx = ((size_t)b * T_DIM + t) * U_DIM + n0 + col;
      out[idx] = tanhf(acc[i] + Xs[b * XS_S + col]);
    }

    grid_barrier(cnt, gen, NBLK_SCAN);
  }
}

// ---------------------------------------------------------------------------
extern "C" void kernel_launch(void* const* d_in, const int* in_sizes, int n_in,
                              void* d_out, int out_size, void* d_ws,
                              size_t ws_size, hipStream_t stream) {
  const float* x = (const float*)d_in[0];    // [B,T,D]
  const float* wax = (const float*)d_in[1];  // [D,U]
  const float* waa = (const float*)d_in[2];  // [U,U]
  const float* ba = (const float*)d_in[3];   // [1,U]
  float* out = (float*)d_out;                // [B,T,U]
  unsigned* bar = (unsigned*)d_ws;

  (void)hipMemsetAsync(d_ws, 0, 256, stream);  // capture-safe barrier reset

  dim3 g1((B_DIM * T_DIM) / 32, U_DIM / 64);
  xp_gemm_kernel<<<g1, dim3(256), 0, stream>>>(x, wax, ba, out);

  const size_t shmem =
      (size_t)(U_DIM * 40 + 32 * 1028 + 32 * 36) * sizeof(float);
  rnn_scan_kernel<<<dim3(NBLK_SCAN), dim3(128), shmem, stream>>>(waa, out,
                                                                 bar);
}